// DiscretizedMixLogisticLoss_1726576854126
// MI455X (gfx1250) — compile-verified
//
#include <hip/hip_runtime.h>

// DiscretizedMixLogisticLoss for MI455X (gfx1250, wave32).
// N=8, C=3, K=10, H=W=256.  l laid out as (N, 4, C, K, H, W); x as (N, C, H, W).
// HBM-bandwidth-bound streaming kernel: single pass over l with 128-bit
// non-temporal loads. All transcendentals in native base-2 (v_exp_f32 /
// v_log_f32 with no conversion multiplies). Reduction over K uses a
// register-resident max (no online rescale):
//   nll = ln2 * ( log2(sum_k 2^{a2_k - m2}) - log2(sum_k 2^{a2_k - m2} p_k) )
// with a2_k = logit_k * log2(e), p_k = clip(cdf_hi - cdf_lo, 1e-12).

#define HW4  16384   // (256*256)/4 float4 groups per plane
#define NCH  3
#define NMIX 10

#define L2E   1.4426950408889634f   // log2(e)
#define C0Q   0.5287663729448977f   // log2(log2(e))
#define LN2   0.6931471805599453f

typedef __attribute__((ext_vector_type(4))) float v4f;

__device__ __forceinline__ float fexp2(float x) {
#if defined(__HIP_DEVICE_COMPILE__) && __has_builtin(__builtin_amdgcn_exp2f)
    return __builtin_amdgcn_exp2f(x);   // v_exp_f32
#else
    return exp2f(x);
#endif
}
__device__ __forceinline__ float flog2(float x) {
#if defined(__HIP_DEVICE_COMPILE__) && __has_builtin(__builtin_amdgcn_logf)
    return __builtin_amdgcn_logf(x);    // v_log_f32 (base-2)
#else
    return log2f(x);
#endif
}
__device__ __forceinline__ float frcp(float x) {
#if defined(__HIP_DEVICE_COMPILE__) && __has_builtin(__builtin_amdgcn_rcpf)
    return __builtin_amdgcn_rcpf(x);    // v_rcp_f32
#else
    return 1.0f / x;
#endif
}

__device__ __forceinline__ v4f v_splat(float s) {
    v4f r; r.x = s; r.y = s; r.z = s; r.w = s; return r;
}
__device__ __forceinline__ v4f v_exp2_4(v4f a) {
    v4f r; r.x = fexp2(a.x); r.y = fexp2(a.y); r.z = fexp2(a.z); r.w = fexp2(a.w); return r;
}
__device__ __forceinline__ v4f v_log2_4(v4f a) {
    v4f r; r.x = flog2(a.x); r.y = flog2(a.y); r.z = flog2(a.z); r.w = flog2(a.w); return r;
}
__device__ __forceinline__ v4f v_rcp4(v4f a) {
    v4f r; r.x = frcp(a.x); r.y = frcp(a.y); r.z = frcp(a.z); r.w = frcp(a.w); return r;
}
// sigmoid(z) = 1 / (1 + exp2(-z*log2e)); rcp maps inf -> 0 for saturation.
__device__ __forceinline__ v4f v_sig4(v4f z) {
    return v_rcp4(v_exp2_4(z * (-L2E)) + 1.0f);
}
__device__ __forceinline__ v4f v_max4(v4f a, v4f b) {
    v4f r; r.x = fmaxf(a.x, b.x); r.y = fmaxf(a.y, b.y);
    r.z = fmaxf(a.z, b.z); r.w = fmaxf(a.w, b.w); return r;
}
__device__ __forceinline__ v4f v_maxs(v4f a, float b) {
    v4f r; r.x = fmaxf(a.x, b); r.y = fmaxf(a.y, b);
    r.z = fmaxf(a.z, b); r.w = fmaxf(a.w, b); return r;
}
__device__ __forceinline__ v4f v_clamp(v4f a, float lo, float hi) {
    v4f r;
    r.x = fminf(fmaxf(a.x, lo), hi); r.y = fminf(fmaxf(a.y, lo), hi);
    r.z = fminf(fmaxf(a.z, lo), hi); r.w = fminf(fmaxf(a.w, lo), hi);
    return r;
}
__device__ __forceinline__ v4f v_ge(v4f a, float b) {   // (a >= b) ? 1 : 0
    v4f r;
    r.x = (a.x >= b) ? 1.0f : 0.0f; r.y = (a.y >= b) ? 1.0f : 0.0f;
    r.z = (a.z >= b) ? 1.0f : 0.0f; r.w = (a.w >= b) ? 1.0f : 0.0f;
    return r;
}
__device__ __forceinline__ v4f v_le(v4f a, float b) {   // (a <= b) ? 1 : 0
    v4f r;
    r.x = (a.x <= b) ? 1.0f : 0.0f; r.y = (a.y <= b) ? 1.0f : 0.0f;
    r.z = (a.z <= b) ? 1.0f : 0.0f; r.w = (a.w <= b) ? 1.0f : 0.0f;
    return r;
}
__device__ __forceinline__ v4f ntload(const v4f* p) {
    return __builtin_nontemporal_load(p);
}

__global__ __launch_bounds__(256)
void dmll_kernel(const float* __restrict__ xg,
                 const float* __restrict__ lg,
                 float* __restrict__ outg,
                 int n_groups)
{
    int g = blockIdx.x * 256 + threadIdx.x;
    if (g >= n_groups) return;
    int n   = g >> 14;        // g / HW4
    int hw4 = g & 16383;      // g % HW4

    const v4f* xv = (const v4f*)xg;
    const v4f* lv = (const v4f*)lg;
    v4f*       ov = (v4f*)outg;

    const int xbase = n * NCH * HW4 + hw4;                 // float4 index into x / out
    const int lbase = n * (4 * NCH * NMIX) * HW4 + hw4;    // float4 index into l

    // x channels (each used by the autoregressive means of later channels)
    v4f x0 = ntload(xv + xbase);
    v4f x1 = ntload(xv + xbase + HW4);
    v4f x2 = ntload(xv + xbase + 2 * HW4);
    v4f xs[3] = {x0, x1, x2};

#pragma unroll
    for (int c = 0; c < NCH; ++c) {
        v4f xc = xs[c];
        v4f lo_cond = v_ge(xc, 0.001f);     // x >= X_LO_BOUND
        v4f hi_cond = v_le(xc, 254.999f);   // x <= X_HI_BOUND
        v4f one_m_hi = 1.0f - hi_cond;

        // Pass 1: all K logits for this channel, pre-scaled to base-2 domain.
        v4f a2[NMIX];
#pragma unroll
        for (int k = 0; k < NMIX; ++k) {
            a2[k] = ntload(lv + lbase + k * HW4 + ((0 * 3 + c) * NMIX) * HW4) * L2E;
        }
        v4f m2 = a2[0];
#pragma unroll
        for (int k = 1; k < NMIX; ++k) m2 = v_max4(m2, a2[k]);

        // Pass 2: accumulate sw = sum 2^(a2-m2), st = sum 2^(a2-m2) * p.
        v4f sw = v_splat(0.0f), st = v_splat(0.0f);
#pragma unroll
        for (int k = 0; k < NMIX; ++k) {
            int bk = lbase + k * HW4;
            v4f mean = ntload(lv + bk + ((1 * 3 + c) * NMIX) * HW4);
            v4f lsc  = ntload(lv + bk + ((2 * 3 + c) * NMIX) * HW4);

            v4f m = mean;
            if (c == 1) {
                v4f c0 = ntload(lv + bk + ((3 * 3 + 0) * NMIX) * HW4);
                m = m + v_sig4(c0) * x0;
            } else if (c == 2) {
                v4f c1 = ntload(lv + bk + ((3 * 3 + 1) * NMIX) * HW4);
                v4f c2 = ntload(lv + bk + ((3 * 3 + 2) * NMIX) * HW4);
                m = m + v_sig4(c1) * x0 + v_sig4(c2) * x1;
            }
            m = v_clamp(m, 0.0f, 255.0f);

            v4f centered = xc - m;
            // q = inv_stdv * log2e = 2^(-clip(ls)*log2e + log2(log2e))
            v4f t = v_maxs(lsc, -7.0f);
            v4f q = v_exp2_4(t * (-L2E) + C0Q);
            v4f u = q * centered;
            v4f d = 0.5f * q;
            // sigmoid(inv*(centered -+ 0.5)) = 1/(1 + 2^(+-d - u))
            v4f elo = v_exp2_4(d - u);
            v4f ehi = v_exp2_4(-(u + d));
            v4f cdf_lo = lo_cond * v_rcp4(elo + 1.0f);
            v4f cdf_hi = hi_cond * v_rcp4(ehi + 1.0f) + one_m_hi;
            v4f p = v_maxs(cdf_hi - cdf_lo, 1e-12f);    // prob mass, clipped

            v4f ea = v_exp2_4(a2[k] - m2);              // <= 1
            sw = sw + ea;
            st = st + ea * p;
        }

        // nll = LSE(logits) - LSE(logits + log p); shared max cancels.
        v4f nll = LN2 * (v_log2_4(sw) - v_log2_4(st));
        __builtin_nontemporal_store(nll, ov + xbase + c * HW4);
    }
}

extern "C" void kernel_launch(void* const* d_in, const int* in_sizes, int n_in,
                              void* d_out, int out_size, void* d_ws, size_t ws_size,
                              hipStream_t stream) {
    const float* x = (const float*)d_in[0];   // (N, C, H, W) fp32
    const float* l = (const float*)d_in[1];   // (N, 4*C*K, H, W) fp32
    float* out = (float*)d_out;               // (N, C, H, W) fp32

    // x has N*C*H*W elements; each thread handles one float4 across all C channels.
    int n_groups = in_sizes[0] / (NCH * 4);   // = N * H * W / 4 = 131072

    dim3 block(256);
    dim3 grid((n_groups + 255) / 256);
    dmll_kernel<<<grid, block, 0, stream>>>(x, l, out, n_groups);
}